// RewardGNN_78855599555029
// MI455X (gfx1250) — compile-verified
//
#include <hip/hip_runtime.h>

typedef __attribute__((ext_vector_type(2))) float v2f;
typedef __attribute__((ext_vector_type(8))) float v8f;

// ---------------------------------------------------------------------------
// C[M,NW] = A[M,K] @ B[K,NW], all fp32, via V_WMMA_F32_16X16X4_F32.
// One wave computes one 16x16 tile of C. M % 16 == 0, K % 4 == 0, NW % 16 == 0.
//
// Fragment layouts per CDNA5 ISA 7.12.2 (wave32):
//  A (16x4):  lane<16 -> M=lane,     K = k0 + {0,1} in v[0],v[1]
//             lane>=16 -> M=lane-16, K = k0 + {2,3}
//  B (4x16):  mirrored: lane-half selects K offset {0,2}, VGPR selects +0/+1,
//             N = n0 + (lane&15)
//  C (16x16): VGPR r: lanes 0-15 -> M=m0+r, lanes 16-31 -> M=m0+8+r; N=n0+(lane&15)
// ---------------------------------------------------------------------------
template <int K, int NW>
__global__ __launch_bounds__(256) void gemm_wmma_f32(
    const float* __restrict__ A, const float* __restrict__ B,
    float* __restrict__ C, int M)
{
    const int lane = threadIdx.x & 31;
    const int wid  = blockIdx.x * (blockDim.x >> 5) + (threadIdx.x >> 5);
    const int tilesN = NW / 16;
    const int tilesM = M / 16;
    if (wid >= tilesM * tilesN) return;
    const int tm = wid / tilesN;
    const int tn = wid - tm * tilesN;
    const int m0 = tm * 16, n0 = tn * 16;
    const int l15  = lane & 15;
    const int koff = (lane >> 4) << 1;              // 0 for lanes 0-15, 2 for 16-31

    const float* arow = A + (size_t)(m0 + l15) * K; // this lane's A row
    const float* bcol = B + (size_t)n0 + l15;       // this lane's B column

    v8f acc = {};
    for (int k0 = 0; k0 < K; k0 += 4) {
        v2f a = *(const v2f*)(arow + k0 + koff);    // K = k0+koff, k0+koff+1
        v2f b;
        b.x = bcol[(size_t)(k0 + koff) * NW];
        b.y = bcol[(size_t)(k0 + koff + 1) * NW];
        acc = __builtin_amdgcn_wmma_f32_16x16x4_f32(
            /*neg_a=*/false, a, /*neg_b=*/false, b,
            /*c_mod=*/(short)0, acc, /*reuse_a=*/false, /*reuse_b=*/false);
    }

    float* crow = C + (size_t)(m0 + (lane >> 4) * 8) * NW + n0 + l15;
#pragma unroll
    for (int r = 0; r < 8; ++r) crow[(size_t)r * NW] = acc[r];
}

// deg[i] = 1.0f  (self-loop weight)
__global__ void k_deg_init(float* deg, int n) {
    int i = blockIdx.x * blockDim.x + threadIdx.x;
    if (i < n) deg[i] = 1.0f;
}

// deg[dst[e]] += ew[e]
__global__ void k_deg_scatter(const long long* __restrict__ dst,
                              const float* __restrict__ ew,
                              float* __restrict__ deg, int e) {
    int i = blockIdx.x * blockDim.x + threadIdx.x;
    if (i < e) atomicAdd(&deg[(int)dst[i]], ew[i]);
}

// dinv[i] = deg>0 ? rsqrt(deg) : 0   (deg >= 1 always here)
__global__ void k_dinv(const float* __restrict__ deg, float* __restrict__ dinv, int n) {
    int i = blockIdx.x * blockDim.x + threadIdx.x;
    if (i < n) {
        float d = deg[i];
        dinv[i] = d > 0.0f ? rsqrtf(d) : 0.0f;
    }
}

// One wave per edge: out[dst] += dinv[src]*ew*dinv[dst] * xw[src]   (F floats)
template <int F>
__global__ __launch_bounds__(256) void k_edge_agg(
    const long long* __restrict__ src, const long long* __restrict__ dst,
    const float* __restrict__ ew, const float* __restrict__ dinv,
    const float* __restrict__ xw, float* __restrict__ out, int e)
{
    const int lane = threadIdx.x & 31;
    const int wid  = blockIdx.x * (blockDim.x >> 5) + (threadIdx.x >> 5);
    if (wid >= e) return;
    const int s = (int)src[wid];
    const int d = (int)dst[wid];
    const float norm = dinv[s] * ew[wid] * dinv[d];
    const float4* xs = (const float4*)(xw + (size_t)s * F);
    float* od = out + (size_t)d * F;
#pragma unroll
    for (int i = lane; i < F / 4; i += 32) {
        float4 v = xs[i];
        atomicAdd(od + 4 * i + 0, norm * v.x);
        atomicAdd(od + 4 * i + 1, norm * v.y);
        atomicAdd(od + 4 * i + 2, norm * v.z);
        atomicAdd(od + 4 * i + 3, norm * v.w);
    }
}

// h = relu(agg + dinv^2 * xw + bias)   (self-loop folded in); in-place ok.
template <int F>
__global__ void k_finish(const float* __restrict__ agg, const float* __restrict__ xw,
                         const float* __restrict__ dinv, const float* __restrict__ bias,
                         float* __restrict__ h, int n)
{
    long long t = (long long)blockIdx.x * blockDim.x + threadIdx.x;
    if (t >= (long long)n * F) return;
    int node = (int)(t >> (F == 128 ? 7 : 6));
    int k    = (int)(t & (F - 1));
    float di = dinv[node];
    float v  = agg[t] + di * di * xw[t] + bias[k];
    h[t] = fmaxf(v, 0.0f);
}

// One wave per node: sums[batch[i]][0..63] += h[i][0..63]; cnt[batch[i]] += 1
__global__ __launch_bounds__(256) void k_pool(
    const float* __restrict__ h, const long long* __restrict__ batch,
    float* __restrict__ sums, float* __restrict__ cnt, int n)
{
    const int lane = threadIdx.x & 31;
    const int wid  = blockIdx.x * (blockDim.x >> 5) + (threadIdx.x >> 5);
    if (wid >= n) return;
    const int g = (int)batch[wid];
    const float* hp = h + (size_t)wid * 64;
    atomicAdd(&sums[g * 64 + lane], hp[lane]);
    atomicAdd(&sums[g * 64 + 32 + lane], hp[32 + lane]);
    if (lane == 0) atomicAdd(&cnt[g], 1.0f);
}

// out[g] = (sums[g]/max(cnt,1)) . Wfc + bfc
__global__ void k_head(const float* __restrict__ sums, const float* __restrict__ cnt,
                       const float* __restrict__ Wfc, const float* __restrict__ bfc,
                       float* __restrict__ out, int g_total)
{
    int g = blockIdx.x * blockDim.x + threadIdx.x;
    if (g >= g_total) return;
    float c = fmaxf(cnt[g], 1.0f);
    float acc = 0.0f;
#pragma unroll
    for (int k = 0; k < 64; ++k) acc += (sums[g * 64 + k] / c) * Wfc[k];
    out[g] = acc + bfc[0];
}

extern "C" void kernel_launch(void* const* d_in, const int* in_sizes, int n_in,
                              void* d_out, int out_size, void* d_ws, size_t ws_size,
                              hipStream_t stream)
{
    const float*     x     = (const float*)d_in[0];      // [N,256]
    const long long* eidx  = (const long long*)d_in[1];  // [2,E]
    const float*     ew    = (const float*)d_in[2];      // [E]
    const long long* batch = (const long long*)d_in[3];  // [N]
    const float*     W1    = (const float*)d_in[4];      // [256,128]
    const float*     b1    = (const float*)d_in[5];      // [128]
    const float*     W2    = (const float*)d_in[6];      // [128,64]
    const float*     b2    = (const float*)d_in[7];      // [64]
    const float*     Wfc   = (const float*)d_in[8];      // [64,1]
    const float*     bfc   = (const float*)d_in[9];      // [1]
    float*           out   = (float*)d_out;              // [G,1]

    const int N = in_sizes[0] / 256;
    const int E = in_sizes[1] / 2;
    const int G = out_size;
    const long long* esrc = eidx;
    const long long* edst = eidx + E;

    // Workspace carve-up (floats). bufA reused: xw1 -> [xw2 | agg2/h2].
    float* ws   = (float*)d_ws;
    float* bufA = ws;                          // N*128
    float* bufB = bufA + (size_t)N * 128;      // N*128  (agg1 -> h1 in place)
    float* deg  = bufB + (size_t)N * 128;      // N
    float* dinv = deg + N;                     // N
    float* sums = dinv + N;                    // G*64
    float* cnt  = sums + (size_t)G * 64;       // G
    (void)ws_size;

    const int T = 256;
    auto cdiv = [](long long a, long long b) { return (int)((a + b - 1) / b); };

    // --- degrees + normalization ---
    k_deg_init<<<cdiv(N, T), T, 0, stream>>>(deg, N);
    k_deg_scatter<<<cdiv(E, T), T, 0, stream>>>(edst, ew, deg, E);
    k_dinv<<<cdiv(N, T), T, 0, stream>>>(deg, dinv, N);

    // --- layer 1: xw1 = x @ W1 (fp32 WMMA) ---
    {
        int tiles = (N / 16) * (128 / 16);
        gemm_wmma_f32<256, 128><<<cdiv(tiles, T / 32), T, 0, stream>>>(x, W1, bufA, N);
    }
    hipMemsetAsync(bufB, 0, (size_t)N * 128 * sizeof(float), stream);
    k_edge_agg<128><<<cdiv((long long)E * 32, T), T, 0, stream>>>(esrc, edst, ew, dinv,
                                                                  bufA, bufB, E);
    k_finish<128><<<cdiv((long long)N * 128, T), T, 0, stream>>>(bufB, bufA, dinv, b1,
                                                                 bufB, N);

    // --- layer 2: xw2 = h1 @ W2 (fp32 WMMA); agg2 lives in bufA's upper half ---
    float* xw2  = bufA;
    float* agg2 = bufA + (size_t)N * 64;
    {
        int tiles = (N / 16) * (64 / 16);
        gemm_wmma_f32<128, 64><<<cdiv(tiles, T / 32), T, 0, stream>>>(bufB, W2, xw2, N);
    }
    hipMemsetAsync(agg2, 0, (size_t)N * 64 * sizeof(float), stream);
    k_edge_agg<64><<<cdiv((long long)E * 32, T), T, 0, stream>>>(esrc, edst, ew, dinv,
                                                                 xw2, agg2, E);
    k_finish<64><<<cdiv((long long)N * 64, T), T, 0, stream>>>(agg2, xw2, dinv, b2,
                                                               agg2, N);

    // --- global mean pool + linear head ---
    hipMemsetAsync(sums, 0, (size_t)(G * 64 + G) * sizeof(float), stream);
    k_pool<<<cdiv((long long)N * 32, T), T, 0, stream>>>(agg2, batch, sums, cnt, N);
    k_head<<<1, G, 0, stream>>>(sums, cnt, Wfc, bfc, out, G);
}